// DBLoss_32074815766649
// MI455X (gfx1250) — compile-verified
//
#include <hip/hip_runtime.h>
#include <math.h>

// ---------------- problem constants (match reference) ----------------
#define BATCH      32
#define NA         3
#define HH         80
#define WW         80
#define NC         80          // NUM_CLASSES
#define CH         85          // 5 + NUM_CLASSES
#define MLAB       20
#define NCELLS     (BATCH*NA*HH*WW)      // 614400
#define BLK        256
#define NBLK       (NCELLS/BLK)          // 2400
#define PADN       2432                  // NBLK padded to multiple of 64
#define NPAD       (PADN-NBLK)           // 32 pad slots per array
#define BLK_PER_B  (NA*HH*WW/BLK)        // 75 blocks per batch image

#define EPSF       1e-7f
#define PI2F       9.869604401089358f    // pi^2
#define STRIDEF    8.0f
#define IMGF       640.0f

__device__ __constant__ float c_aw[3] = {10.0f/8.0f, 16.0f/8.0f, 33.0f/8.0f};
__device__ __constant__ float c_ah[3] = {13.0f/8.0f, 30.0f/8.0f, 23.0f/8.0f};

typedef __attribute__((ext_vector_type(2))) float v2f;
typedef __attribute__((ext_vector_type(8))) float v8f;

// focal BCE on a logit l vs target t (ALPHA=0.25, GAMMA=1.5)
__device__ __forceinline__ float focal_bce(float l, float t) {
    float bce = fmaxf(l, 0.f) - l * t + log1pf(expf(-fabsf(l)));
    float p   = 1.f / (1.f + expf(-l));
    float pt  = p * t + (1.f - p) * (1.f - t);
    float x   = fminf(fmaxf(1.f - pt, 0.f), 1.f);
    return 0.25f * (x * sqrtf(x)) * bce;   // x^1.5
}

// ---------------- main per-cell loss kernel (gather form) ----------------
__global__ __launch_bounds__(BLK)
void yolo_loss_main(const float* __restrict__ p_raw,
                    const float* __restrict__ labels,        // (B,M,5)
                    const unsigned char* __restrict__ lmask, // (B,M) bool8
                    const float* __restrict__ clsw,          // (NC)
                    float* __restrict__ partials)            // 4*PADN floats
{
    __shared__ int   sILo[MLAB], sIHi[MLAB], sJLo[MLAB], sJHi[MLAB];
    __shared__ int   sBa[MLAB],  sCls[MLAB];
    __shared__ float sGx[MLAB], sGy[MLAB], sGw[MLAB], sGh[MLAB];
    __shared__ float red0[BLK], red1[BLK], red2[BLK], red3[BLK];

    const int t   = threadIdx.x;
    const int bid = blockIdx.x;
    const int b   = bid / BLK_PER_B;                 // whole block is one batch image

    // --- preprocess this image's labels into LDS (replaces assign_targets scatter) ---
    if (t < MLAB) {
        const float* L = labels + (size_t)(b * MLAB + t) * 5;
        int   cls = (int)L[0];
        float gx = L[1] * IMGF, gy = L[2] * IMGF, gw = L[3] * IMGF, gh = L[4] * IMGF;
        int gi = (int)fminf(fmaxf(gx / STRIDEF, 0.f), (float)WW - 0.001f);
        int gj = (int)fminf(fmaxf(gy / STRIDEF, 0.f), (float)HH - 0.001f);
        float gtw = gw / STRIDEF, gth = gh / STRIDEF;
        // best anchor by IoU ratio; argmax ties -> first index (strict >)
        int   ba = 0; float best = -1.f;
        #pragma unroll
        for (int a = 0; a < 3; ++a) {
            float inter = fminf(gtw, c_aw[a]) * fminf(gth, c_ah[a]);
            float uni   = gtw * gth + c_aw[a] * c_ah[a] - inter + 1e-9f;
            float r = inter / uni;
            if (r > best) { best = r; ba = a; }
        }
        bool valid = lmask[b * MLAB + t] != 0;
        sBa[t]  = valid ? ba : -1;                    // -1: never matches
        sCls[t] = cls;
        sILo[t] = max(gi - 1, 0);  sIHi[t] = min(gi + 1, WW - 1);
        sJLo[t] = max(gj - 1, 0);  sJHi[t] = min(gj + 1, HH - 1);
        sGx[t] = gx; sGy[t] = gy; sGw[t] = gw; sGh[t] = gh;
    }
    __syncthreads();

    // --- decode this thread's cell ---
    const int idx  = bid * BLK + t;                  // < NCELLS exactly
    int rem  = idx - b * (NA * HH * WW);
    int a    = rem / (HH * WW);
    int rem2 = rem - a * (HH * WW);
    int j    = rem2 / WW;
    int i    = rem2 - j * WW;

    // --- gather targets: is this cell positive? which classes? which box? ---
    unsigned int cbits0 = 0u, cbits1 = 0u, cbits2 = 0u;
    float tgx = 0.f, tgy = 0.f, tgw = 0.f, tgh = 0.f;
    bool pos = false;
    #pragma unroll 4
    for (int m = 0; m < MLAB; ++m) {
        bool hit = (a == sBa[m]) &&
                   (i >= sILo[m]) && (i <= sIHi[m]) &&
                   (j >= sJLo[m]) && (j <= sJHi[m]);
        if (hit) {
            pos = true;
            int c = sCls[m];
            if (c < 32)       cbits0 |= 1u << c;
            else if (c < 64)  cbits1 |= 1u << (c - 32);
            else              cbits2 |= 1u << (c - 64);
            tgx = sGx[m]; tgy = sGy[m]; tgw = sGw[m]; tgh = sGh[m];
        }
    }

    const float* cell = p_raw + (size_t)idx * CH;

    // objectness focal BCE (every cell reads only channel 4: 1 load / 1 line)
    float tObj  = pos ? 1.f : 0.f;
    float accO  = focal_bce(cell[4], tObj);
    float accB  = 0.f, accC = 0.f, accN = 0.f;

    if (pos) {
        // warm the class-logit cachelines early (gfx1250 global_prefetch_b8)
        __builtin_prefetch((const void*)(cell + 5),  0, 1);
        __builtin_prefetch((const void*)(cell + 37), 0, 1);
        __builtin_prefetch((const void*)(cell + 69), 0, 1);

        accN = 1.f;
        // ---- CIoU box loss ----
        float px = ((float)i + 1.f / (1.f + expf(-cell[0]))) * STRIDEF;
        float py = ((float)j + 1.f / (1.f + expf(-cell[1]))) * STRIDEF;
        float pw = expf(cell[2]) * c_aw[a] * STRIDEF;
        float ph = expf(cell[3]) * c_ah[a] * STRIDEF;

        float px1 = px - pw * 0.5f, py1 = py - ph * 0.5f;
        float px2 = px + pw * 0.5f, py2 = py + ph * 0.5f;
        float gx1 = tgx - tgw * 0.5f, gy1 = tgy - tgh * 0.5f;
        float gx2 = tgx + tgw * 0.5f, gy2 = tgy + tgh * 0.5f;

        float iw = fmaxf(fminf(px2, gx2) - fmaxf(px1, gx1), 0.f);
        float ih = fmaxf(fminf(py2, gy2) - fmaxf(py1, gy1), 0.f);
        float inter = iw * ih;
        float uni = fmaxf(px2 - px1, 0.f) * fmaxf(py2 - py1, 0.f)
                  + fmaxf(gx2 - gx1, 0.f) * fmaxf(gy2 - gy1, 0.f)
                  - inter + EPSF;
        float iou = inter / uni;
        float cw = fmaxf(fmaxf(px2, gx2) - fminf(px1, gx1), 0.f);
        float chh = fmaxf(fmaxf(py2, gy2) - fminf(py1, gy1), 0.f);
        float c2 = cw * cw + chh * chh + EPSF;
        float rho2 = (px - tgx) * (px - tgx) + (py - tgy) * (py - tgy);
        float dat = atanf(tgw / (tgh + EPSF)) - atanf(pw / (ph + EPSF));
        float v = (4.f / PI2F) * dat * dat;
        float alpha = v / (1.f - iou + v + EPSF);
        accB = 1.f - (iou - rho2 / c2 - alpha * v);

        // ---- class focal BCE, only for positive cells ----
        #pragma unroll 8
        for (int c = 0; c < NC; ++c) {
            unsigned int bit = (c < 32) ? ((cbits0 >> c) & 1u)
                             : (c < 64) ? ((cbits1 >> (c - 32)) & 1u)
                                        : ((cbits2 >> (c - 64)) & 1u);
            accC += focal_bce(cell[5 + c], (float)bit) * clsw[c];
        }
    }

    // --- block tree reduction (wave32-friendly, 256 threads = 8 waves) ---
    red0[t] = accB; red1[t] = accO; red2[t] = accC; red3[t] = accN;
    __syncthreads();
    for (int s = BLK / 2; s > 0; s >>= 1) {
        if (t < s) {
            red0[t] += red0[t + s];
            red1[t] += red1[t + s];
            red2[t] += red2[t + s];
            red3[t] += red3[t + s];
        }
        __syncthreads();
    }
    if (t == 0) {
        partials[0 * PADN + bid] = red0[0];
        partials[1 * PADN + bid] = red1[0];
        partials[2 * PADN + bid] = red2[0];
        partials[3 * PADN + bid] = red3[0];
    }
    // first NPAD(=32) blocks also zero the pad tail so the WMMA fold needs no
    // per-lane masking (keeps EXEC all-1s and loads unconditional)
    if (t == 1 && bid < NPAD) {
        #pragma unroll
        for (int arr = 0; arr < 4; ++arr)
            partials[arr * PADN + NBLK + bid] = 0.f;
    }
}

// ---- fold PADN fp32 partials with the matrix pipe: D = A(16x4) * ones(4x16) + C ----
// A layout (ISA 7.12.2, 32-bit A 16x4): lanes 0-15 hold (M=lane, K=0/1),
// lanes 16-31 hold (M=lane-16, K=2/3). Element (m,k) -> src[m*4+k]; m*4+k0 is
// always even, so each lane issues one aligned global_load_b64 per iteration.
// Each WMMA issue folds 64 values; C chains as the accumulator (exact f32 FMA).
__device__ __forceinline__ float wmma_fold(const float* __restrict__ src, int lane)
{
    v8f c = {};
    v2f ones; ones[0] = 1.f; ones[1] = 1.f;
    const int m  = lane & 15;
    const int k0 = (lane >> 4) << 1;          // 0 for lanes 0-15, 2 for 16-31
    const float* p = src + m * 4 + k0;
    #pragma unroll 2
    for (int base = 0; base < PADN; base += 64) {   // 38 unconditional iterations
        v2f av = *(const v2f*)(p + base);
        c = __builtin_amdgcn_wmma_f32_16x16x4_f32(
                false, av, false, ones, (short)0, c, false, false);
    }
    // D[m][n] = rowsum(m) for all n; VGPR r: lanes0-15 -> M=r, lanes16-31 -> M=8+r
    float s = c[0] + c[1] + c[2] + c[3] + c[4] + c[5] + c[6] + c[7];
    return __shfl(s, 0, 32) + __shfl(s, 16, 32);   // rows 0-7 + rows 8-15
}

__global__ __launch_bounds__(32)
void yolo_loss_reduce(const float* __restrict__ partials, float* __restrict__ out)
{
    const int lane = threadIdx.x;                 // one wave32, EXEC all-1s
    float sBox = wmma_fold(partials + 0 * PADN, lane);
    float sObj = wmma_fold(partials + 1 * PADN, lane);
    float sCls = wmma_fold(partials + 2 * PADN, lane);
    float sNp  = wmma_fold(partials + 3 * PADN, lane);
    if (lane == 0) {
        float np = fmaxf(sNp, 1.f);
        float loss = 7.5f * (sBox / np)
                   + 1.0f * (sObj * (1.0f / (float)NCELLS))
                   + 0.5f * (sCls / (np * (float)NC));
        out[0] = loss;
    }
}

extern "C" void kernel_launch(void* const* d_in, const int* in_sizes, int n_in,
                              void* d_out, int out_size, void* d_ws, size_t ws_size,
                              hipStream_t stream)
{
    const float*         p_raw  = (const float*)d_in[0];
    const float*         labels = (const float*)d_in[1];
    const unsigned char* lmask  = (const unsigned char*)d_in[2]; // bool8
    const float*         clsw   = (const float*)d_in[3];
    float*               out    = (float*)d_out;
    float*               part   = (float*)d_ws;   // 4*PADN floats (~39 KB)

    yolo_loss_main<<<NBLK, BLK, 0, stream>>>(p_raw, labels, lmask, clsw, part);
    yolo_loss_reduce<<<1, 32, 0, stream>>>(part, out);
}